// GCNLayer_39565238731446
// MI455X (gfx1250) — compile-verified
//
#include <hip/hip_runtime.h>

typedef __attribute__((ext_vector_type(2))) float v2f;
typedef __attribute__((ext_vector_type(8))) float v8f;

#define BM 128
#define BN 128
#define BK 16
#define LDA 20   // stride for A / W tiles: mult of 4 (b128 align), conflict-free
#define LDB 18   // stride for transposed S tile: even (b64 align), conflict-free

#if __has_builtin(__builtin_amdgcn_global_load_async_to_lds_b128)
#define ASYNC_LDS 1
#else
#define ASYNC_LDS 0
#endif

#if ASYNC_LDS
typedef int v4i_t __attribute__((vector_size(16)));
typedef v4i_t __attribute__((address_space(1)))* gv4i_p;
typedef v4i_t __attribute__((address_space(3)))* lv4i_p;
#endif

__device__ __forceinline__ void async_cp16(const float* g, float* l) {
#if ASYNC_LDS
    __builtin_amdgcn_global_load_async_to_lds_b128(
        (gv4i_p)(const void*)g, (lv4i_p)(void*)l, 0, 0);
#endif
}

__device__ __forceinline__ void wait_async0() {
#if ASYNC_LDS
#if __has_builtin(__builtin_amdgcn_s_wait_asynccnt)
    __builtin_amdgcn_s_wait_asynccnt(0);
#else
    asm volatile("s_wait_asynccnt 0x0" ::: "memory");
#endif
#endif
}

__device__ __forceinline__ v8f wmma_f32(v2f a, v2f b, v8f c) {
    return __builtin_amdgcn_wmma_f32_16x16x4_f32(
        false, a, false, b, (short)0, c, false, false);
}

// ---------------------------------------------------------------- deg/dis ---
__global__ __launch_bounds__(256)
void deg_kernel(const float* __restrict__ adj, float* __restrict__ dis, int N) {
    const int wave = (blockIdx.x * 256 + threadIdx.x) >> 5;
    const int lane = threadIdx.x & 31;
    const float* p = adj + (size_t)wave * N;
    float s = 0.f;
    #pragma unroll 4
    for (int c = lane * 4; c < N; c += 128) {
        float4 v = *(const float4*)(p + c);
        s += v.x + v.y + v.z + v.w;
    }
    #pragma unroll
    for (int off = 16; off > 0; off >>= 1) s += __shfl_xor(s, off, 32);
    if (lane == 0) dis[wave] = rsqrtf(s + 1.0f);
}

// ------------------------------------------------- linear: S = dis*(XW^T+b) --
__global__ __launch_bounds__(256)
void lin_kernel(const float* __restrict__ X, const float* __restrict__ W,
                const float* __restrict__ bias, const float* __restrict__ dis,
                float* __restrict__ S, int K, int Nout) {
    __shared__ float As[2][BM * LDA];
    __shared__ float Bs[2][BN * LDA];
    const int tid = threadIdx.x;
    const int lane = tid & 31, w = tid >> 5;
    const int wm = w & 3, wn = w >> 2;
    const int m0 = blockIdx.y * BM, n0 = blockIdx.x * BN;
    const int lq = lane >> 4, lr = lane & 15;
    const int rA0 = tid >> 2, cA = (tid & 3) << 2;   // 128x16 tile: 2 chunks/thr

    v8f acc[2][4];
    #pragma unroll
    for (int mt = 0; mt < 2; ++mt)
        #pragma unroll
        for (int nt = 0; nt < 4; ++nt) acc[mt][nt] = 0;

#if !ASYNC_LDS
    float4 aReg[2], wReg[2];
#endif

    // issue + commit tile 0
    #pragma unroll
    for (int i = 0; i < 2; ++i) {
        const int r = rA0 + i * 64;
#if ASYNC_LDS
        async_cp16(X + (size_t)(m0 + r) * K + cA, &As[0][r * LDA + cA]);
        async_cp16(W + (size_t)(n0 + r) * K + cA, &Bs[0][r * LDA + cA]);
#else
        aReg[i] = *(const float4*)(X + (size_t)(m0 + r) * K + cA);
        wReg[i] = *(const float4*)(W + (size_t)(n0 + r) * K + cA);
#endif
    }
#if !ASYNC_LDS
    #pragma unroll
    for (int i = 0; i < 2; ++i) {
        const int r = rA0 + i * 64;
        *(float4*)&As[0][r * LDA + cA] = aReg[i];
        *(float4*)&Bs[0][r * LDA + cA] = wReg[i];
    }
#endif
    wait_async0();
    __syncthreads();

    const int T = K / BK;
    for (int t = 0; t < T; ++t) {
        const int cb = t & 1, nb = cb ^ 1;
        if (t + 1 < T) {
            const int k0 = (t + 1) * BK;
            #pragma unroll
            for (int i = 0; i < 2; ++i) {
                const int r = rA0 + i * 64;
#if ASYNC_LDS
                async_cp16(X + (size_t)(m0 + r) * K + k0 + cA, &As[nb][r * LDA + cA]);
                async_cp16(W + (size_t)(n0 + r) * K + k0 + cA, &Bs[nb][r * LDA + cA]);
#else
                aReg[i] = *(const float4*)(X + (size_t)(m0 + r) * K + k0 + cA);
                wReg[i] = *(const float4*)(W + (size_t)(n0 + r) * K + k0 + cA);
#endif
            }
        }
        const float* Ab = As[cb];
        const float* Bb = Bs[cb];
        #pragma unroll
        for (int kk = 0; kk < BK; kk += 4) {
            const int kc = kk + 2 * lq;
            v2f a[2], bb[4];
            #pragma unroll
            for (int mt = 0; mt < 2; ++mt)
                a[mt] = *(const v2f*)&Ab[(wm * 32 + mt * 16 + lr) * LDA + kc];
            #pragma unroll
            for (int nt = 0; nt < 4; ++nt)
                bb[nt] = *(const v2f*)&Bb[(wn * 64 + nt * 16 + lr) * LDA + kc];
            #pragma unroll
            for (int mt = 0; mt < 2; ++mt)
                #pragma unroll
                for (int nt = 0; nt < 4; ++nt)
                    acc[mt][nt] = wmma_f32(a[mt], bb[nt], acc[mt][nt]);
        }
        if (t + 1 < T) {
#if !ASYNC_LDS
            #pragma unroll
            for (int i = 0; i < 2; ++i) {
                const int r = rA0 + i * 64;
                *(float4*)&As[nb][r * LDA + cA] = aReg[i];
                *(float4*)&Bs[nb][r * LDA + cA] = wReg[i];
            }
#endif
            wait_async0();
        }
        __syncthreads();
    }

    #pragma unroll
    for (int mt = 0; mt < 2; ++mt)
        #pragma unroll
        for (int nt = 0; nt < 4; ++nt) {
            const int n = n0 + wn * 64 + nt * 16 + lr;
            #pragma unroll
            for (int r = 0; r < 8; ++r) {
                const int m = m0 + wm * 32 + mt * 16 + r + 8 * lq;
                S[(size_t)m * Nout + n] = dis[m] * (acc[mt][nt][r] + bias[n]);
            }
        }
}

// ------------------------------- propagation: out = dis[m]*(adj@S + S[m,:]) --
__global__ __launch_bounds__(256)
void prop_kernel(const float* __restrict__ adj, const float* __restrict__ S,
                 const float* __restrict__ dis, float* __restrict__ out,
                 int N, int H) {
    __shared__ float As[2][BM * LDA];
    __shared__ float Bt[2][BN * LDB];
    const int tid = threadIdx.x;
    const int lane = tid & 31, w = tid >> 5;
    const int wm = w & 3, wn = w >> 2;
    const int b = blockIdx.z, m0 = blockIdx.y * BM, n0 = blockIdx.x * BN;
    const int lq = lane >> 4, lr = lane & 15;
    const float* Ap = adj + (size_t)b * N * N;
    const float* Sp = S + (size_t)b * N * H;
    const int rA0 = tid >> 2, cA = (tid & 3) << 2;   // A: 128x16, 2 chunks/thr
    const int rB0 = tid >> 5, cB = (tid & 31) << 2;  // B: 16x128,  2 chunks/thr

    v8f acc[2][4];
    #pragma unroll
    for (int mt = 0; mt < 2; ++mt)
        #pragma unroll
        for (int nt = 0; nt < 4; ++nt) acc[mt][nt] = 0;

#if !ASYNC_LDS
    float4 aReg[2];
#endif
    float4 bReg[2];

    // issue + commit tile 0
    #pragma unroll
    for (int i = 0; i < 2; ++i) {
        const int r = rA0 + i * 64;
#if ASYNC_LDS
        async_cp16(Ap + (size_t)(m0 + r) * N + cA, &As[0][r * LDA + cA]);
#else
        aReg[i] = *(const float4*)(Ap + (size_t)(m0 + r) * N + cA);
#endif
        bReg[i] = *(const float4*)(Sp + (size_t)(rB0 + i * 8) * H + n0 + cB);
    }
    #pragma unroll
    for (int i = 0; i < 2; ++i) {
        const int kr = rB0 + i * 8;
        Bt[0][(cB + 0) * LDB + kr] = bReg[i].x;
        Bt[0][(cB + 1) * LDB + kr] = bReg[i].y;
        Bt[0][(cB + 2) * LDB + kr] = bReg[i].z;
        Bt[0][(cB + 3) * LDB + kr] = bReg[i].w;
#if !ASYNC_LDS
        const int r = rA0 + i * 64;
        *(float4*)&As[0][r * LDA + cA] = aReg[i];
#endif
    }
    wait_async0();
    __syncthreads();

    const int T = N / BK;
    for (int t = 0; t < T; ++t) {
        const int cb = t & 1, nb = cb ^ 1;
        if (t + 1 < T) {
            const int k0 = (t + 1) * BK;
            #pragma unroll
            for (int i = 0; i < 2; ++i) {
                const int r = rA0 + i * 64;
#if ASYNC_LDS
                async_cp16(Ap + (size_t)(m0 + r) * N + k0 + cA, &As[nb][r * LDA + cA]);
#else
                aReg[i] = *(const float4*)(Ap + (size_t)(m0 + r) * N + k0 + cA);
#endif
                bReg[i] = *(const float4*)(Sp + (size_t)(k0 + rB0 + i * 8) * H + n0 + cB);
            }
        }
        const float* Ab = As[cb];
        const float* Bb = Bt[cb];
        #pragma unroll
        for (int kk = 0; kk < BK; kk += 4) {
            const int kc = kk + 2 * lq;
            v2f a[2], bb[4];
            #pragma unroll
            for (int mt = 0; mt < 2; ++mt)
                a[mt] = *(const v2f*)&Ab[(wm * 32 + mt * 16 + lr) * LDA + kc];
            #pragma unroll
            for (int nt = 0; nt < 4; ++nt)
                bb[nt] = *(const v2f*)&Bb[(wn * 64 + nt * 16 + lr) * LDB + kc];
            #pragma unroll
            for (int mt = 0; mt < 2; ++mt)
                #pragma unroll
                for (int nt = 0; nt < 4; ++nt)
                    acc[mt][nt] = wmma_f32(a[mt], bb[nt], acc[mt][nt]);
        }
        if (t + 1 < T) {
            #pragma unroll
            for (int i = 0; i < 2; ++i) {
                const int kr = rB0 + i * 8;
                Bt[nb][(cB + 0) * LDB + kr] = bReg[i].x;
                Bt[nb][(cB + 1) * LDB + kr] = bReg[i].y;
                Bt[nb][(cB + 2) * LDB + kr] = bReg[i].z;
                Bt[nb][(cB + 3) * LDB + kr] = bReg[i].w;
#if !ASYNC_LDS
                const int r = rA0 + i * 64;
                *(float4*)&As[nb][r * LDA + cA] = aReg[i];
#endif
            }
            wait_async0();
        }
        __syncthreads();
    }

    #pragma unroll
    for (int mt = 0; mt < 2; ++mt)
        #pragma unroll
        for (int nt = 0; nt < 4; ++nt) {
            const int n = n0 + wn * 64 + nt * 16 + lr;
            #pragma unroll
            for (int r = 0; r < 8; ++r) {
                const int m = m0 + wm * 32 + mt * 16 + r + 8 * lq;
                const float sv = Sp[(size_t)m * H + n];   // +I term
                out[((size_t)b * N + m) * H + n] =
                    dis[b * N + m] * (acc[mt][nt][r] + sv);
            }
        }
}

// ------------------------------------------------------------- BN statistics -
__global__ __launch_bounds__(512)
void zero_kernel(float* __restrict__ sums) { sums[threadIdx.x] = 0.f; }

__global__ __launch_bounds__(256)
void stats_kernel(const float* __restrict__ x, float* __restrict__ sums,
                  int rows, int H) {
    const int tid = threadIdx.x;
    const int chunk = rows / gridDim.x;
    const int r0 = blockIdx.x * chunk;
    float s = 0.f, s2 = 0.f;
    for (int r = 0; r < chunk; ++r) {
        float v = x[(size_t)(r0 + r) * H + tid];
        s += v; s2 += v * v;
    }
    atomicAdd(&sums[tid], s);
    atomicAdd(&sums[H + tid], s2);
}

// ------------------------------------------- BN apply + ReLU + residual ------
__global__ __launch_bounds__(256)
void bn_kernel(float* __restrict__ out, const float* __restrict__ h,
               const float* __restrict__ sums, const float* __restrict__ bnw,
               const float* __restrict__ bnb, const float* __restrict__ rw,
               int H, float invM) {
    const size_t i = (size_t)blockIdx.x * 256 + threadIdx.x;
    const int hh = (int)(i & (size_t)(H - 1));
    const float mean = sums[hh] * invM;
    const float var  = sums[H + hh] * invM - mean * mean;
    const float y = (out[i] - mean) * rsqrtf(var + 1e-5f) * bnw[hh] + bnb[hh];
    out[i] = fmaxf(y, 0.f) + rw[0] * h[i];
}

extern "C" void kernel_launch(void* const* d_in, const int* in_sizes, int n_in,
                              void* d_out, int out_size, void* d_ws, size_t ws_size,
                              hipStream_t stream) {
    const float* h   = (const float*)d_in[0];
    const float* adj = (const float*)d_in[1];
    const float* W   = (const float*)d_in[2];
    const float* bv  = (const float*)d_in[3];
    const float* bnw = (const float*)d_in[4];
    const float* bnb = (const float*)d_in[5];
    const float* rw  = (const float*)d_in[6];
    float* out = (float*)d_out;

    const int B = 16, N = 2048, H = 256;
    float* ws   = (float*)d_ws;
    float* dis  = ws;                       // B*N floats
    float* S    = ws + (size_t)B * N;       // B*N*H floats
    float* sums = S + (size_t)B * N * H;    // 2*H floats

    deg_kernel<<<(B * N) / 8, 256, 0, stream>>>(adj, dis, N);

    dim3 g2(H / BN, (B * N) / BM);
    lin_kernel<<<g2, 256, 0, stream>>>(h, W, bv, dis, S, H, H);

    zero_kernel<<<1, 2 * H, 0, stream>>>(sums);

    dim3 g3(H / BN, N / BM, B);
    prop_kernel<<<g3, 256, 0, stream>>>(adj, S, dis, out, N, H);

    stats_kernel<<<128, 256, 0, stream>>>(out, sums, B * N, H);

    bn_kernel<<<(int)(((size_t)B * N * H) / 256), 256, 0, stream>>>(
        out, h, sums, bnw, bnb, rw, H, 1.0f / (float)(B * N));
}